// VectorQuantizer_10660108828679
// MI455X (gfx1250) — compile-verified
//
#include <hip/hip_runtime.h>

typedef __attribute__((ext_vector_type(2))) float v2f;
typedef __attribute__((ext_vector_type(8))) float v8f;
typedef __attribute__((ext_vector_type(4))) unsigned int u32x4;
typedef __attribute__((ext_vector_type(8))) int i32x8;
typedef __attribute__((ext_vector_type(4))) int i32x4;

#define N_E   1024
#define DIM   256
#define HW    1024          // 32*32 spatial
#define NPTS  32768         // 32*32*32 points
#define ROWS_PER_WAVE 16
#define ROWS_PER_WG   128   // 8 waves * 16 rows
#define CODES_PER_PASS 16   // one 16-wide WMMA M-tile per pass, double buffered
#define NPASS (N_E / CODES_PER_PASS)
#define EMB_STRIDE 260      // 256 + 4 pad: conflict-free ds_load_b64 (TDM pad reproduces this)
#define LOSS_SCALE (1.25f / ((float)NPTS * (float)DIM))

#if defined(__gfx1250__) && __has_builtin(__builtin_amdgcn_tensor_load_to_lds) && \
    __has_builtin(__builtin_amdgcn_s_wait_tensorcnt)
#define USE_TDM 1
#else
#define USE_TDM 0
#endif

// ---------------- helper kernels ----------------

__global__ void vq_init(float* out) {
    if (threadIdx.x == 0) out[NPTS] = 0.0f;   // zero the loss slot
}

// ||e||^2 per codebook row -> workspace
__global__ void vq_enorm(const float* __restrict__ emb, float* __restrict__ enorm) {
    int e = blockIdx.x * blockDim.x + threadIdx.x;
    if (e < N_E) {
        const float* r = emb + (size_t)e * DIM;
        float s = 0.0f;
        #pragma unroll 8
        for (int c = 0; c < DIM; ++c) s += r[c] * r[c];
        enorm[e] = s;
    }
}

// sum(z^2) * scale accumulated into the loss slot (layout-independent)
__global__ void vq_znorm(const float* __restrict__ z, float* __restrict__ out) {
    __shared__ float red[256];
    const size_t total = (size_t)NPTS * DIM;
    size_t i = (size_t)blockIdx.x * blockDim.x + threadIdx.x;
    const size_t step = (size_t)gridDim.x * blockDim.x;
    float s = 0.0f;
    for (; i < total; i += step) { float v = z[i]; s += v * v; }
    red[threadIdx.x] = s;
    __syncthreads();
    for (int off = 128; off > 0; off >>= 1) {
        if ((int)threadIdx.x < off) red[threadIdx.x] += red[threadIdx.x + off];
        __syncthreads();
    }
    if (threadIdx.x == 0) atomicAdd(out + NPTS, red[0] * LOSS_SCALE);
}

// ---------------- TDM staging of one 16x256-f32 codebook panel ----------------
#if USE_TDM
__device__ static inline void tdm_stage(const float* emb, int tile, void* ldsDst) {
    // LDS byte offset of destination buffer (AS(3) ptrtoint yields LDS offset)
    unsigned lds_off = (unsigned)(unsigned long long)
        (__attribute__((address_space(3))) char*)ldsDst;
    unsigned long long ga =
        (unsigned long long)(uintptr_t)(emb + (size_t)tile * CODES_PER_PASS * DIM);

    // D# group 0: count=1 | lds_addr | global_addr[56:0] | type=2
    u32x4 g0 = { 1u, lds_off, (unsigned)ga,
                 ((unsigned)(ga >> 32) & 0x1FFFFFFu) | (2u << 30) };
    // D# group 1:
    //  d0: data_size=4B (2<<16) | pad_enable (1<<20) | pad_interval=7 (256 dw)
    //      | pad_amount=3 (4 dw)  -> LDS row stride 260 floats
    //  tensor_dim0=256, tensor_dim1=16, tile_dim0=256, tile_dim1=16,
    //  tensor_dim0_stride=256, tensor_dim1_stride=256
    i32x8 g1 = { (int)0x07D20000,
                 (int)0x01000000,   // [63:48] tensor_dim0 lo16 = 256
                 (int)0x00100000,   // [95:80] tensor_dim1 lo16 = 16
                 (int)0x01000000,   // [127:112] tile_dim0 = 256
                 16,                // [143:128] tile_dim1 = 16
                 256,               // [191:160] tensor_dim0_stride lo32
                 (int)0x01000000,   // [223:208] tensor_dim1_stride lo16 = 256
                 0 };
    i32x4 g2 = { 0, 0, 0, 0 };      // 2D tensor: groups 2/3 unused
    i32x4 g3 = { 0, 0, 0, 0 };
#if defined(__clang_major__) && (__clang_major__ >= 23)
    i32x8 g4 = { 0, 0, 0, 0, 0, 0, 0, 0 };
    __builtin_amdgcn_tensor_load_to_lds(g0, g1, g2, g3, g4, 0);
#else
    __builtin_amdgcn_tensor_load_to_lds(g0, g1, g2, g3, 0);
#endif
}
#endif

// ---------------- fused GEMM + argmin ----------------
// D = E (16 codes x K) x Z^T (K x 16 rows), fp32 WMMA 16x16x4.
// Running per-lane argmin of (||e||^2 - 2 z.e); ||z||^2 folded in via vq_znorm.

__global__ __launch_bounds__(256) void vq_main(const float* __restrict__ z,
                                               const float* __restrict__ emb,
                                               const float* __restrict__ enorm,
                                               float* __restrict__ out) {
    __shared__ float embT[2][CODES_PER_PASS * EMB_STRIDE]; // 2 x 16640 B

    const int tid  = threadIdx.x;
    const int wave = tid >> 5;
    const int lane = tid & 31;
    const int l16  = lane & 15;   // row (N) / code (M) position in tile
    const int h    = lane >> 4;   // half-wave select

    const int rowBase = blockIdx.x * ROWS_PER_WG + wave * ROWS_PER_WAVE;
    const int b  = rowBase >> 10;        // batch index (HW = 1024)
    const int hw = rowBase & 1023;
    const float* zbase = z + (size_t)b * DIM * HW + hw + l16 + (size_t)(2 * h) * HW;

    float vmin[8];
    int   vidx[8];
    #pragma unroll
    for (int r = 0; r < 8; ++r) { vmin[r] = 3.4e38f; vidx[r] = 0; }

#if USE_TDM
    if (wave == 0) tdm_stage(emb, 0, &embT[0][0]);   // prologue: stage pass 0
#endif

    for (int pass = 0; pass < NPASS; ++pass) {
        const int cur = pass & 1;
#if USE_TDM
        if (wave == 0) {
            if (pass + 1 < NPASS) {
                tdm_stage(emb, pass + 1, &embT[cur ^ 1][0]);
                __builtin_amdgcn_s_wait_tensorcnt(1);  // in-order: pass's panel landed
            } else {
                __builtin_amdgcn_s_wait_tensorcnt(0);
            }
        }
        __syncthreads();                               // panel visible to all waves
#else
        __syncthreads();
        {   // cooperative fallback staging: 16 rows x 256 f32
            const int code = tid >> 4;          // 0..15
            const int cseg = (tid & 15) * 16;   // 0..240
            const float4* src = (const float4*)(emb +
                (size_t)(pass * CODES_PER_PASS + code) * DIM + cseg);
            float* dst = &embT[cur][0] + code * EMB_STRIDE + cseg;
            #pragma unroll
            for (int j = 0; j < 4; ++j) ((float4*)dst)[j] = src[j];
        }
        __syncthreads();
#endif

        v8f acc = {};
        const float* e0 = &embT[cur][0] + l16 * EMB_STRIDE + 2 * h;

        #pragma unroll 8
        for (int k = 0; k < DIM / 4; ++k) {
            v2f a;  a.x  = e0[4 * k];  a.y  = e0[4 * k + 1];
            v2f bb; bb.x = zbase[(size_t)(4 * k) * HW];
                    bb.y = zbase[(size_t)(4 * k + 1) * HW];
            acc = __builtin_amdgcn_wmma_f32_16x16x4_f32(
                false, a, false, bb, (short)0, acc, false, false);
        }

        // lane holds codes m = 8h + r for its row l16; two float4 of ||e||^2
        const float4 en0 = *(const float4*)(enorm + pass * CODES_PER_PASS + 8 * h);
        const float4 en1 = *(const float4*)(enorm + pass * CODES_PER_PASS + 8 * h + 4);
        const float en[8] = { en0.x, en0.y, en0.z, en0.w, en1.x, en1.y, en1.z, en1.w };
        const int cb = pass * CODES_PER_PASS + 8 * h;
        #pragma unroll
        for (int r = 0; r < 8; ++r) {
            float d = en[r] - 2.0f * acc[r];
            if (d < vmin[r]) { vmin[r] = d; vidx[r] = cb + r; }   // strict '<': lowest idx wins
        }
#if USE_TDM
        __syncthreads();   // all reads of this buffer done before TDM overwrites it
#endif
    }

    // local argmin over this lane's 8 code slots
    float bv = vmin[0]; int bi = vidx[0];
    #pragma unroll
    for (int r = 1; r < 8; ++r)
        if (vmin[r] < bv || (vmin[r] == bv && vidx[r] < bi)) { bv = vmin[r]; bi = vidx[r]; }

    // merge with partner half-wave (same row, other 8 mod-16 code residues)
    float pv = __shfl_xor(bv, 16, 32);
    int   pi = __shfl_xor(bi, 16, 32);
    if (pv < bv || (pv == bv && pi < bi)) { bv = pv; bi = pi; }

    const int row = rowBase + l16;
    if (h == 0) out[row] = (float)bi;

    // loss: sum over the wave's 16 rows of min(||e||^2 - 2 z.e)
    float s = (h == 0) ? bv : 0.0f;
    #pragma unroll
    for (int off = 16; off > 0; off >>= 1) s += __shfl_xor(s, off, 32);
    if (lane == 0) atomicAdd(out + NPTS, s * LOSS_SCALE);
}

// ---------------- host entry ----------------

extern "C" void kernel_launch(void* const* d_in, const int* in_sizes, int n_in,
                              void* d_out, int out_size, void* d_ws, size_t ws_size,
                              hipStream_t stream) {
    const float* z   = (const float*)d_in[0];   // (32,256,32,32) fp32
    const float* emb = (const float*)d_in[1];   // (1024,256) fp32
    float* out   = (float*)d_out;               // 32768 idx (as float) + 1 loss
    float* enorm = (float*)d_ws;                // 1024 floats scratch

    vq_init  <<<1,   1,   0, stream>>>(out);
    vq_enorm <<<4,   256, 0, stream>>>(emb, enorm);
    vq_znorm <<<256, 256, 0, stream>>>(z, out);
    vq_main  <<<NPTS / ROWS_PER_WG, 256, 0, stream>>>(z, emb, enorm, out);
}